// NormStability_25434796327759
// MI455X (gfx1250) — compile-verified
//
#include <hip/hip_runtime.h>
#include <math.h>

// NormStability for MI455X (gfx1250, wave32).
//
// Reference: per-element loop over p in [0,128]; only p in [P_low, P_high]
// contributes, and P_high - P_low <= 8 (THRESHOLD*L = 6.4). best_len depends
// ONLY on the integer pair (P_low, width) -> 129*9 = 1161 distinct values.
// Strategy: build the 1161-entry max_stab LUT once per block in LDS
// (4.6 KB of the 320 KB/WGP), then stream elements with float4 (b128) loads,
// one ds_load gather + one IEEE divide per element. Memory floor:
// 12 MB / 23.3 TB/s ~ 0.5 us; this formulation is close to it.

#define LBITS 7
#define LVAL  128                 // L = stream_len = 128 (power of two), len_f == L
#define WMAX  8                   // max (P_high - P_low): ceil(0.05*128)+1 bound
#define LUT_W (WMAX + 1)
#define LUT_SIZE ((LVAL + 1) * LUT_W)

// Exact re-implementation of the reference fori_loop restricted to the valid
// window [plow, phigh]; returns max_stab = max(1 - best_len/L, 0).
__device__ __forceinline__ float max_stab_for(int plow, int phigh) {
    const float Lf  = (float)LVAL;
    const float Plo = (float)plow;
    const float Phi = (float)phigh;
    float best = Lf;                                  // init best_len = L_f
    for (int p = plow; p <= phigh; ++p) {
        // l_p = L / gcd(p, L); L = 2^7 so gcd = 2^min(ctz(p),7); gcd(0,L)=L.
        int tz = (p == 0) ? LBITS : __builtin_ctz(p);
        if (tz > LBITS) tz = LBITS;
        const float l_p    = (float)(LVAL >> tz);
        const float inv_lp = 1.0f / l_p;              // exact: power of two
        const float pf     = (float)p;

        // ---- R_for(B = 0) ----
        float R_low0  = (p == plow)
                          ? ((Plo <= 0.0f) ? 1.0f : Lf)
                          : (Plo * inv_lp) / (pf - Plo);          // (Plo-0)/l_p/d_low
        float R_high0 = (p == phigh)
                          ? 1.0f                                   // 0 <= Phi always
                          : ((0.0f - Phi) * inv_lp) / (Phi - pf);  // (0-Phi)/l_p/d_high
        float R0 = ceilf(fmaxf(R_low0, R_high0));

        // ---- R_for(B = L) ----
        float R_lowL  = (p == plow)
                          ? 1.0f                                   // Plo <= L always
                          : ((Plo - Lf) * inv_lp) / (pf - Plo);
        float R_highL = (p == phigh)
                          ? ((Lf <= Phi) ? 1.0f : Lf)
                          : ((Lf - Phi) * inv_lp) / (Phi - pf);
        float RL = ceilf(fmaxf(R_lowL, R_highL));

        float R    = fminf(R0, RL);
        float cand = R * l_p;
        if (cand < best) best = fmaxf(cand, 1.0f);
    }
    return fmaxf(1.0f - best * (1.0f / (float)LVAL), 0.0f);
}

__device__ __forceinline__ float norm_stab_elem(float v, float sl, const float* lut) {
    // bipolar window around (v+1)/2, THRESHOLD*0.5 = 0.025
    float a        = (v + 1.0f) * 0.5f;
    float min_prob = fmaxf(a - 0.025f, 0.0f);
    float max_prob = fminf(a + 0.025f, 1.0f);
    int plow  = (int)fminf(fmaxf(floorf(min_prob * 128.0f), 0.0f), 128.0f);
    int phigh = (int)fminf(fmaxf(ceilf (max_prob * 128.0f), 0.0f), 128.0f);
    int w = phigh - plow;
    if (w > WMAX) w = WMAX;                            // mathematically <= 8
    float max_stab  = lut[plow * LUT_W + w];           // ds_load gather
    float stability = 1.0f - fminf(fmaxf(sl, 1.0f), 128.0f) * (1.0f / 128.0f);
    float ns = stability / max_stab;                   // 0/0 -> NaN, x/0 -> inf
    ns = (ns != ns) ? 0.0f : ns;                       // NaN -> 0 (reference)
    return fminf(fmaxf(ns, 0.0f), 1.0f);               // clip to [0,1]
}

__global__ void __launch_bounds__(256)
NormStability_25434796327759_kernel(const float* __restrict__ in_value,
                                    const float* __restrict__ stable_len,
                                    float* __restrict__ out,
                                    int n, int n4) {
    __shared__ float lut[LUT_SIZE];
    // Cooperative LUT build: ~4.5 entries/thread, <=9 window iterations each.
    for (int i = threadIdx.x; i < LUT_SIZE; i += blockDim.x) {
        int plow  = i / LUT_W;
        int w     = i - plow * LUT_W;
        int phigh = plow + w;
        if (phigh > LVAL) phigh = LVAL;                // unreachable entries, harmless
        lut[i] = max_stab_for(plow, phigh);
    }
    __syncthreads();

    const float4* __restrict__ v4 = (const float4*)in_value;
    const float4* __restrict__ s4 = (const float4*)stable_len;
    float4* __restrict__       o4 = (float4*)out;

    const int stride = (int)(gridDim.x * blockDim.x);
    for (int i = (int)(blockIdx.x * blockDim.x + threadIdx.x); i < n4; i += stride) {
        int nxt = i + stride;                          // gfx1250 global_prefetch_b8
        if (nxt < n4) {
            __builtin_prefetch(&v4[nxt], 0, 1);
            __builtin_prefetch(&s4[nxt], 0, 1);
        }
        float4 v = v4[i];                              // global_load_b128
        float4 s = s4[i];
        float4 r;
        r.x = norm_stab_elem(v.x, s.x, lut);
        r.y = norm_stab_elem(v.y, s.y, lut);
        r.z = norm_stab_elem(v.z, s.z, lut);
        r.w = norm_stab_elem(v.w, s.w, lut);
        o4[i] = r;                                     // global_store_b128
    }

    // Scalar tail (n is 2^20 in practice so this is empty, kept for safety).
    for (int i = n4 * 4 + (int)(blockIdx.x * blockDim.x + threadIdx.x); i < n; i += stride) {
        out[i] = norm_stab_elem(in_value[i], stable_len[i], lut);
    }
}

extern "C" void kernel_launch(void* const* d_in, const int* in_sizes, int n_in,
                              void* d_out, int out_size, void* d_ws, size_t ws_size,
                              hipStream_t stream) {
    const float* in_value   = (const float*)d_in[0];
    const float* stable_len = (const float*)d_in[1];
    float*       out        = (float*)d_out;
    // d_in[2] is the static Python scalar stream_len (=128); baked in as LVAL.

    int n  = in_sizes[0];
    int n4 = n / 4;

    const int block = 256;                 // 8 wave32 waves per workgroup
    int grid = 512;                        // grid-stride: amortize LDS LUT build
    int needed = (n4 + block - 1) / block;
    if (needed == 0) needed = 1;
    if (grid > needed) grid = needed;

    NormStability_25434796327759_kernel<<<grid, block, 0, stream>>>(
        in_value, stable_len, out, n, n4);
}